// QLSTM_65481071408802
// MI455X (gfx1250) — compile-verified
//
#include <hip/hip_runtime.h>
#include <math.h>

#define T_STEPS 512
#define BATCH   256
#define D_INP   128
#define D_HID   128
#define NQ      8
#define DCOMB   256
#define NG      32   // 4 gates * NQ outputs

typedef __attribute__((ext_vector_type(2))) float v2f;
typedef __attribute__((ext_vector_type(8))) float v8f;

__device__ __forceinline__ v8f wmma_f32_4(v2f a, v2f b, v8f c) {
  // D = A(16x4 f32) * B(4x16 f32) + C(16x16 f32)
  return __builtin_amdgcn_wmma_f32_16x16x4_f32(
      /*neg_a=*/false, a, /*neg_b=*/false, b,
      /*c_mod=*/(short)0, c, /*reuse_a=*/false, /*reuse_b=*/false);
}

#define WAIT_ASYNC() asm volatile("s_wait_asynccnt 0" ::: "memory")

// Async global->LDS copy of one 32KB ZX timestep slab (2048 x b128 chunks,
// 2 per thread with 1024 threads). Tracked by ASYNCcnt; overlapped with the
// whole step's compute and drained at the end-of-step barrier.
__device__ __forceinline__ void zx_prefetch(const float* __restrict__ ZX, int t,
                                            float* dstbuf, int tid) {
  const float* src = ZX + (size_t)t * (BATCH * NG);
  const unsigned base = (unsigned)(uintptr_t)dstbuf;   // LDS byte offset
#pragma unroll
  for (int h = 0; h < 2; ++h) {
    const int c = tid + h * 1024;                      // 16-byte chunk id
    const unsigned long long ga =
        (unsigned long long)(uintptr_t)(src + c * 4);
    const unsigned lds = base + (unsigned)(c * 16);
    asm volatile("global_load_async_to_lds_b128 %0, %1, off"
                 :: "v"(lds), "v"(ga) : "memory");
  }
}

// ---------------------------------------------------------------------------
// Kernel A: ZX[t*B + b, g] = sum_k X[t,b,k] * Wx_all[g,k]
// Wx_all[g, :] = W_{gate}[q, 0:128], gate = g>>3, q = g&7.
// One wave per 16-row M tile, both 16-col N tiles per wave.
// ---------------------------------------------------------------------------
__global__ __launch_bounds__(256)
void qlstm_zx_gemm(const float* __restrict__ X,
                   const float* __restrict__ Wf, const float* __restrict__ Wi,
                   const float* __restrict__ Wu, const float* __restrict__ Wo,
                   float* __restrict__ ZX) {
  __shared__ float s_w[NG * D_INP];   // [32][128] row-major (Wx part)
  const int tid = threadIdx.x;
  const float* Wg[4] = {Wf, Wi, Wu, Wo};
  for (int i = tid; i < NG * D_INP; i += 256) {
    int g = i >> 7, col = i & 127;
    s_w[i] = Wg[g >> 3][(g & 7) * DCOMB + col];
  }
  __syncthreads();

  const int wave  = tid >> 5;
  const int lane  = tid & 31;
  const int mt    = blockIdx.x * 8 + wave;     // 0 .. T*B/16 - 1
  const int row0  = mt * 16;
  const int mA    = lane & 15;
  const int khalf = (lane >> 4) * 2;           // K half select per ISA layout
  const int nn    = lane & 15;

  const float* arow = X + (size_t)(row0 + mA) * D_INP;

  v8f acc0 = {}; v8f acc1 = {};
  for (int ks = 0; ks < D_INP / 4; ++ks) {
    const int k = ks * 4 + khalf;
    v2f a; a.x = arow[k]; a.y = arow[k + 1];
    v2f b0; b0.x = s_w[nn * D_INP + k];        b0.y = s_w[nn * D_INP + k + 1];
    v2f b1; b1.x = s_w[(nn + 16) * D_INP + k]; b1.y = s_w[(nn + 16) * D_INP + k + 1];
    acc0 = wmma_f32_4(a, b0, acc0);
    acc1 = wmma_f32_4(a, b1, acc1);
  }
#pragma unroll
  for (int r = 0; r < 8; ++r) {
    const int mm = r + ((lane >> 4) << 3);
    const size_t base = (size_t)(row0 + mm) * NG;
    ZX[base + nn]      = acc0[r];
    ZX[base + 16 + nn] = acc1[r];
  }
}

// ---------------------------------------------------------------------------
// Kernel B: sequential scan, single workgroup (32 waves), state in LDS.
//   z = ZX[t] + hx @ Wh_all^T + bias      (WMMA, 16 Mtiles x 2 Ntiles)
//   gate reduction: prod cos(z[:,1:8]) -> sigmoid/tanh
//   cx = f*cx + i*g;  hx = o*tanh(cx);  outs[t] = hx
// ZX[t] is double-buffered in LDS via async global->LDS copies so the L2
// round-trip never sits on the 512-step serial dependency chain.
// ---------------------------------------------------------------------------
__global__ __launch_bounds__(1024)
void qlstm_scan(const float* __restrict__ ZX,
                const float* __restrict__ Wf, const float* __restrict__ Wi,
                const float* __restrict__ Wu, const float* __restrict__ Wo,
                const float* __restrict__ bf, const float* __restrict__ bi,
                const float* __restrict__ bu, const float* __restrict__ bo,
                float* __restrict__ out) {
  extern __shared__ float smem[];
  float* s_h    = smem;                     // [256][128]  hx state
  float* s_w    = s_h + BATCH * D_HID;      // [32][128]   Wh_all
  float* s_z    = s_w + NG * D_HID;         // [256][32]   per-step z
  float* s_gate = s_z + BATCH * NG;         // [256][4]    f,i,g,o
  float* s_bias = s_gate + BATCH * 4;       // [32]
  float* s_zx0  = s_bias + NG;              // [256][32]   ZX double buffer 0
  float* s_zx1  = s_zx0 + BATCH * NG;       // [256][32]   ZX double buffer 1

  const int tid = threadIdx.x;
  const float* Wg[4] = {Wf, Wi, Wu, Wo};
  const float* bg[4] = {bf, bi, bu, bo};
  for (int i = tid; i < NG * D_HID; i += 1024) {
    int g = i >> 7, col = i & 127;
    s_w[i] = Wg[g >> 3][(g & 7) * DCOMB + D_INP + col];   // recurrent half
  }
  if (tid < NG) s_bias[tid] = bg[tid >> 3][tid & 7];
  for (int i = tid; i < BATCH * D_HID; i += 1024) s_h[i] = 0.f;

  float cx[32];
#pragma unroll
  for (int j = 0; j < 32; ++j) cx[j] = 0.f;

  // preload ZX[0] into buffer 0
  zx_prefetch(ZX, 0, s_zx0, tid);
  WAIT_ASYNC();
  __syncthreads();

  // GEMM role: wave -> (mtile, ntile)
  const int wave  = tid >> 5, lane = tid & 31;
  const int mt    = wave & 15, nt = wave >> 4;
  const int row0  = mt * 16;
  const int mA    = lane & 15;
  const int khalf = (lane >> 4) * 2;
  const int nn    = lane & 15;
  // elementwise role: thread -> (row, gate) and (row, 32-wide column slab)
  const int brow  = tid >> 2;
  const int gsel  = tid & 3;
  const int dbase = (tid & 3) * 32;

  for (int t = 0; t < T_STEPS; ++t) {
    float* s_cur = (t & 1) ? s_zx1 : s_zx0;
    float* s_nxt = (t & 1) ? s_zx0 : s_zx1;
    // kick off next step's ZX slab; drains at this step's final barrier
    if (t + 1 < T_STEPS) zx_prefetch(ZX, t + 1, s_nxt, tid);

    // ---- recurrent GEMM into z ----
    v8f acc;
#pragma unroll
    for (int r = 0; r < 8; ++r) {
      const int mm = r + ((lane >> 4) << 3);
      acc[r] = s_cur[(row0 + mm) * NG + nt * 16 + nn] + s_bias[nt * 16 + nn];
    }
    const float* ah = s_h + (size_t)(row0 + mA) * D_HID;
    const float* bw = s_w + (size_t)(nt * 16 + nn) * D_HID;
    for (int ks = 0; ks < D_HID / 4; ++ks) {
      const int k = ks * 4 + khalf;
      v2f a; a.x = ah[k]; a.y = ah[k + 1];
      v2f b; b.x = bw[k]; b.y = bw[k + 1];
      acc = wmma_f32_4(a, b, acc);
    }
#pragma unroll
    for (int r = 0; r < 8; ++r) {
      const int mm = r + ((lane >> 4) << 3);
      s_z[(row0 + mm) * NG + nt * 16 + nn] = acc[r];
    }
    __syncthreads();

    // ---- gate reduction: one (row, gate) per thread ----
    {
      const float* zrow = s_z + brow * NG + gsel * NQ;
      float p = 1.f;
#pragma unroll
      for (int q = 1; q < NQ; ++q) p *= __cosf(zrow[q]);
      float gv = (gsel == 2) ? tanhf(p) : (1.f / (1.f + __expf(-p)));
      s_gate[brow * 4 + gsel] = gv;
    }
    __syncthreads();

    // ---- state update: 4 threads x 32 columns per row ----
    {
      const float f  = s_gate[brow * 4 + 0];
      const float iv = s_gate[brow * 4 + 1];
      const float g  = s_gate[brow * 4 + 2];
      const float o  = s_gate[brow * 4 + 3];
      float* hrow = s_h + brow * D_HID + dbase;
      float* orow = out + ((size_t)t * BATCH + brow) * D_HID + dbase;
#pragma unroll
      for (int j = 0; j < 32; ++j) {
        const float c2 = f * cx[j] + iv * g;
        cx[j] = c2;
        const float h2 = o * tanhf(c2);
        hrow[j] = h2;
        orow[j] = h2;
      }
    }
    WAIT_ASYNC();        // next step's ZX slab resident in LDS
    __syncthreads();     // s_h and s_nxt stable before next step's GEMM
  }

  // ---- final (hx, cx) ----
  {
    float* hx_out = out + (size_t)T_STEPS * BATCH * D_HID;
    float* cx_out = hx_out + BATCH * D_HID;
    for (int i = tid; i < BATCH * D_HID; i += 1024) hx_out[i] = s_h[i];
    float* crow = cx_out + brow * D_HID + dbase;
#pragma unroll
    for (int j = 0; j < 32; ++j) crow[j] = cx[j];
  }
}

// ---------------------------------------------------------------------------
extern "C" void kernel_launch(void* const* d_in, const int* in_sizes, int n_in,
                              void* d_out, int out_size, void* d_ws, size_t ws_size,
                              hipStream_t stream) {
  const float* X  = (const float*)d_in[0];
  const float* Wf = (const float*)d_in[1];
  const float* bf = (const float*)d_in[2];
  const float* Wi = (const float*)d_in[3];
  const float* bi = (const float*)d_in[4];
  const float* Wu = (const float*)d_in[5];
  const float* bu = (const float*)d_in[6];
  const float* Wo = (const float*)d_in[7];
  const float* bo = (const float*)d_in[8];
  // d_in[9..12] (qwf..qwo) drop out analytically (diagonal RZ ops).
  float* out = (float*)d_out;
  float* ZX  = (float*)d_ws;   // T*B*NG floats = 16 MB scratch

  // Phase 1: input projection for all timesteps (parallel, bandwidth-bound).
  const int mtiles = (T_STEPS * BATCH) / 16;         // 8192
  qlstm_zx_gemm<<<mtiles / 8, 256, 0, stream>>>(X, Wf, Wi, Wu, Wo, ZX);

  // Phase 2: latency-bound recurrent scan, single WGP, state in LDS,
  // ZX double-buffered via async global->LDS (ASYNCcnt).
  const size_t shmem = (size_t)(BATCH * D_HID + NG * D_HID + BATCH * NG +
                                BATCH * 4 + NG + 2 * BATCH * NG) * sizeof(float);
  qlstm_scan<<<1, 1024, shmem, stream>>>(ZX, Wf, Wi, Wu, Wo,
                                         bf, bi, bu, bo, out);
}